// ElasticTransformLayer_82875688943836
// MI455X (gfx1250) — compile-verified
//
#include <hip/hip_runtime.h>
#include <math.h>

typedef __attribute__((ext_vector_type(2))) float v2f;
typedef __attribute__((ext_vector_type(8))) float v8f;
typedef __attribute__((ext_vector_type(4))) int   v4i;
typedef __attribute__((address_space(3))) const float* lds_cfp;

#define TH 32
#define TW 32
#define NTHREADS 128
#define LDS_FLOATS 14700   // 57.4 KB staged source region per workgroup

// ---------- gfx1250 async global->LDS helpers ----------
__device__ __forceinline__ unsigned lds_off_u32(const void* p) {
  // generic pointer to __shared__ carries the LDS byte offset in its low 32 bits
  return (unsigned)(unsigned long long)p;
}

__device__ __forceinline__ void async_g2l_b128(const float* g, unsigned loff) {
#if defined(__AMDGCN__)
#if __has_builtin(__builtin_amdgcn_global_load_async_to_lds_b128)
  __builtin_amdgcn_global_load_async_to_lds_b128(
      (__attribute__((address_space(1))) v4i*)(unsigned long long)g,
      (__attribute__((address_space(3))) v4i*)(unsigned long long)loff,
      0, 0);
#else
  unsigned long long ga = (unsigned long long)g;
  asm volatile("global_load_async_to_lds_b128 %0, %1, off"
               :: "v"(loff), "v"(ga) : "memory");
#endif
#else
  (void)g; (void)loff;
#endif
}

__device__ __forceinline__ void wait_async0() {
#if defined(__AMDGCN__)
#if __has_builtin(__builtin_amdgcn_s_wait_asynccnt)
  __builtin_amdgcn_s_wait_asynccnt(0);
#else
  asm volatile("s_wait_asynccnt 0" ::: "memory");
#endif
#endif
}

// ---------- bilinear control-grid weights: linspace(0, 2, n_out) ----------
__device__ __forceinline__ void lin3(int pos, float s, float* w) {
  float u = (float)pos * s;            // in [0, 2]
  int i0 = (int)u; i0 = min(i0, 1);    // clip(floor(u), 0, 1), u >= 0
  float f = u - (float)i0;
  w[0] = (i0 == 0) ? (1.0f - f) : 0.0f;
  w[1] = (i0 == 0) ? f : (1.0f - f);
  w[2] = (i0 == 0) ? 0.0f : f;
}

// ---------- pass 1: global max reduction ----------
__global__ void init_ws_kernel(float* ws) { ws[0] = -3.402823466e38f; }

__device__ __forceinline__ void atomicMaxF(float* a, float v) {
  unsigned* ua = (unsigned*)a;
  unsigned cur = *ua;
  while (__uint_as_float(cur) < v) {
    unsigned prev = atomicCAS(ua, cur, __float_as_uint(v));
    if (prev == cur) break;
    cur = prev;
  }
}

__global__ void __launch_bounds__(256)
maxred_kernel(const float* __restrict__ in, int n, float* __restrict__ ws) {
  const float4* in4 = reinterpret_cast<const float4*>(in);
  int n4 = n >> 2;
  float m = -3.402823466e38f;
  for (int i = blockIdx.x * blockDim.x + threadIdx.x; i < n4;
       i += gridDim.x * blockDim.x) {
    float4 v = in4[i];
    m = fmaxf(m, fmaxf(fmaxf(v.x, v.y), fmaxf(v.z, v.w)));
  }
  if (blockIdx.x == 0 && threadIdx.x == 0) {
    for (int i = n4 << 2; i < n; ++i) m = fmaxf(m, in[i]);
  }
  // wave32 reduction
  for (int off = 16; off > 0; off >>= 1) m = fmaxf(m, __shfl_xor(m, off, 32));
  __shared__ float sm[8];
  if ((threadIdx.x & 31) == 0) sm[threadIdx.x >> 5] = m;
  __syncthreads();
  if (threadIdx.x == 0) {
    float bm = sm[0];
    for (int w = 1; w < (int)(blockDim.x >> 5); ++w) bm = fmaxf(bm, sm[w]);
    atomicMaxF(ws, bm);
  }
}

// ---------- bilinear sample + store, specialized for LDS vs global taps ----------
template <bool USE_LDS>
__device__ __forceinline__ void sample_store(
    const float* __restrict__ in, float* __restrict__ out, lds_cfp t3,
    int H, int W, int rowLo, int colLo, int RW3, float cval,
    int gy0, int gx0, int hl, int nn, const v8f& dyv, const v8f& dxv) {
#pragma unroll
  for (int e = 0; e < 8; ++e) {
    const int gy = gy0 + e + 8 * hl;   // WMMA C layout: row = vgpr + 8*lanehalf
    const int gx = gx0 + nn;           // col = lane & 15
    float cyf = (float)gy + dyv[e];
    float cxf = (float)gx + dxv[e];
    float y0f = floorf(cyf), x0f = floorf(cxf);
    float fy = cyf - y0f, fx = cxf - x0f;
    int y0 = (int)y0f, x0 = (int)x0f;
    float w00 = (1.0f - fy) * (1.0f - fx);
    float w01 = (1.0f - fy) * fx;
    float w10 = fy * (1.0f - fx);
    float w11 = fy * fx;
    float accr = 0.0f, accg = 0.0f, accb = 0.0f;
#pragma unroll
    for (int t = 0; t < 4; ++t) {
      int yi = y0 + (t >> 1);
      int xi = x0 + (t & 1);
      float wt = (t == 0) ? w00 : (t == 1) ? w01 : (t == 2) ? w10 : w11;
      bool valid = (yi >= 0) && (yi < H) && (xi >= 0) && (xi < W);
      int yc = min(max(yi, 0), H - 1);
      int xc = min(max(xi, 0), W - 1);
      float r, g, b;
      if (USE_LDS) {
        int idx = (yc - rowLo) * RW3 + (xc - colLo) * 3;
        r = t3[idx]; g = t3[idx + 1]; b = t3[idx + 2];   // ds_load path
      } else {
        const float* p = in + ((size_t)yc * W + xc) * 3;
        r = p[0]; g = p[1]; b = p[2];
      }
      if (!valid) { r = cval; g = cval; b = cval; }
      accr += wt * r; accg += wt * g; accb += wt * b;
    }
    if (gy < H && gx < W) {
      size_t o = ((size_t)gy * W + gx) * 3;
      out[o] = accr; out[o + 1] = accg; out[o + 2] = accb;
    }
  }
}

// ---------- pass 2: elastic transform ----------
__global__ void __launch_bounds__(NTHREADS)
elastic_kernel(const float* __restrict__ in, const float* __restrict__ disp,
               const float* __restrict__ wsmax, float* __restrict__ out,
               int H, int W) {
  __shared__ float tile[LDS_FLOATS];

  const int tid  = threadIdx.x;
  const int lane = tid & 31;
  const int wv   = tid >> 5;
  const int hl   = lane >> 4;       // lane half
  const int nn   = lane & 15;       // column within 16x16 subtile
  const int ty0  = blockIdx.y * TH;
  const int tx0  = blockIdx.x * TW;
  const int gy0  = ty0 + (wv >> 1) * 16;   // subtile origin (4 waves -> 2x2 subtiles)
  const int gx0  = tx0 + (wv & 1) * 16;
  const float sy = 2.0f / (float)(H - 1);
  const float sx = 2.0f / (float)(W - 1);
  const float cval = wsmax[0];

  // scaled 3x3 control grids (uniform across block)
  float D0[9], D1[9];
#pragma unroll
  for (int i = 0; i < 9; ++i) { D0[i] = disp[i] * 5.0f; D1[i] = disp[9 + i] * 5.0f; }

  // dense field is a convex combination of control values -> tile source bounds
  float d0min = D0[0], d0max = D0[0], d1min = D1[0], d1max = D1[0];
#pragma unroll
  for (int i = 1; i < 9; ++i) {
    d0min = fminf(d0min, D0[i]); d0max = fmaxf(d0max, D0[i]);
    d1min = fminf(d1min, D1[i]); d1max = fmaxf(d1max, D1[i]);
  }
  int rowLo = ty0 + (int)floorf(d0min);
  int rowHi = ty0 + TH - 1 + (int)floorf(d0max) + 1;
  int colLo = tx0 + (int)floorf(d1min);
  int colHi = tx0 + TW - 1 + (int)floorf(d1max) + 1;
  rowLo = min(max(rowLo, 0), H - 1); rowHi = min(max(rowHi, 0), H - 1);
  colLo = min(max(colLo, 0), W - 1); colHi = min(max(colHi, 0), W - 1);
  colLo &= ~3;                                   // 16B-align global row starts
  int RH = rowHi - rowLo + 1;
  int RW = ((colHi - colLo + 1) + 3) & ~3;       // pad to multiple of 4 cols
  const int RW3 = RW * 3;                        // floats per staged row (mult of 4)
  const bool useLds = (RH * RW3) <= LDS_FLOATS;

  // stage source region into LDS with async DMA (overlaps WMMA below)
  if (useLds) {
    const unsigned lbase = lds_off_u32(&tile[0]);
    const int RWv = RW3 >> 2;
    const int nvec = RH * RWv;
    for (int t = tid; t < nvec; t += NTHREADS) {
      int r = t / RWv;
      int c = (t - r * RWv) << 2;
      const float* g = in + ((size_t)(rowLo + r) * W + colLo) * 3 + c;
      async_g2l_b128(g, lbase + (unsigned)((r * RW3 + c) << 2));
    }
  }

  // dense-field einsum via WMMA: dy = Wy[16x3] * (D0 * Wx^T)[3x16]  (K padded to 4)
  float wyv[3]; lin3(gy0 + nn, sy, wyv);   // A rows: M = lane&15
  float wxv[3]; lin3(gx0 + nn, sx, wxv);   // B cols: N = lane&15
  float m0[3], m1[3];
#pragma unroll
  for (int k = 0; k < 3; ++k) {
    m0[k] = D0[k * 3] * wxv[0] + D0[k * 3 + 1] * wxv[1] + D0[k * 3 + 2] * wxv[2];
    m1[k] = D1[k * 3] * wxv[0] + D1[k * 3 + 1] * wxv[1] + D1[k * 3 + 2] * wxv[2];
  }
  v8f dyv, dxv;
#if defined(__AMDGCN__) && __has_builtin(__builtin_amdgcn_wmma_f32_16x16x4_f32)
  {
    // 16x4 A layout: vgpr0/1 = K0/K1 (lanes 0-15), K2/K3 (lanes 16-31)
    v2f a, b0, b1;
    a.x  = hl ? wyv[2] : wyv[0];  a.y  = hl ? 0.0f : wyv[1];
    b0.x = hl ? m0[2]  : m0[0];   b0.y = hl ? 0.0f : m0[1];
    b1.x = hl ? m1[2]  : m1[0];   b1.y = hl ? 0.0f : m1[1];
    v8f z = {0.0f, 0.0f, 0.0f, 0.0f, 0.0f, 0.0f, 0.0f, 0.0f};
    dyv = __builtin_amdgcn_wmma_f32_16x16x4_f32(false, a, false, b0, (short)0, z, false, false);
    dxv = __builtin_amdgcn_wmma_f32_16x16x4_f32(false, a, false, b1, (short)0, z, false, false);
  }
#else
#pragma unroll
  for (int e = 0; e < 8; ++e) {
    int gy = gy0 + e + 8 * hl;
    float wy2[3]; lin3(gy, sy, wy2);
    dyv[e] = wy2[0] * m0[0] + wy2[1] * m0[1] + wy2[2] * m0[2];
    dxv[e] = wy2[0] * m1[0] + wy2[1] * m1[1] + wy2[2] * m1[2];
  }
#endif

  wait_async0();
  __syncthreads();

  // sample 8 pixels per lane; separate instantiations keep LDS taps on ds_load
  lds_cfp t3 = (lds_cfp)(unsigned)(unsigned long long)&tile[0];
  if (useLds) {
    sample_store<true >(in, out, t3, H, W, rowLo, colLo, RW3, cval,
                        gy0, gx0, hl, nn, dyv, dxv);
  } else {
    sample_store<false>(in, out, t3, H, W, rowLo, colLo, RW3, cval,
                        gy0, gx0, hl, nn, dyv, dxv);
  }
}

extern "C" void kernel_launch(void* const* d_in, const int* in_sizes, int n_in,
                              void* d_out, int out_size, void* d_ws, size_t ws_size,
                              hipStream_t stream) {
  const float* in   = (const float*)d_in[0];
  const float* disp = (const float*)d_in[1];
  float* out = (float*)d_out;
  float* ws  = (float*)d_ws;
  const int H = 2048, W = 4096;     // reference shape
  const int n = in_sizes[0];        // H*W*3

  init_ws_kernel<<<1, 1, 0, stream>>>(ws);
  maxred_kernel<<<2048, 256, 0, stream>>>(in, n, ws);

  dim3 grid(W / TW, H / TH);
  elastic_kernel<<<grid, NTHREADS, 0, stream>>>(in, disp, ws, out, H, W);
  (void)n_in; (void)out_size; (void)ws_size;
}